// Impeller_14499809591534
// MI455X (gfx1250) — compile-verified
//
#include <hip/hip_runtime.h>

#define NND   20000
#define IND   256
#define HID   128
#define OUTD  64
#define NLAY  4
#define NPATH 8
#define PLEN  8
#define TILE  32   // node rows per block (two 16-row WMMA tiles sharing B)

// LDS row strides padded so stride % 64 == 4 -> 16 row-lanes hit distinct banks
#define FSTRIDE 260   // fout tile (32 x 256)
#define ASTRIDE 260   // fc_in A tile (32 x 256)
#define BSTRIDE 132   // fc_out A tile (32 x 128)

typedef __attribute__((ext_vector_type(2))) float v2f;
typedef __attribute__((ext_vector_type(8))) float v8f;

// ---------------------------------------------------------------------------
// fc_in: in_feats = relu(X @ Win^T + b)   X:(N,256)  Win:(128,256)
// 1 block = 32 rows x 128 cols, 8 waves; each wave: two 16x16 tiles, shared B.
// ---------------------------------------------------------------------------
__global__ __launch_bounds__(256) void fcin_kernel(
    const float* __restrict__ x, const float* __restrict__ w,
    const float* __restrict__ bias, float* __restrict__ outf)
{
    __shared__ float aLDS[TILE * ASTRIDE];
    const int tid = threadIdx.x;
    const int n0  = blockIdx.x * TILE;

    // stage 32x256 A tile (8192 floats, 8 float4 per thread), coalesced
    for (int i = tid * 4; i < TILE * IND; i += 256 * 4) {
        int r = i >> 8, ccol = i & 255;
        *(float4*)&aLDS[r * ASTRIDE + ccol] =
            *(const float4*)&x[(size_t)(n0 + r) * IND + ccol];
    }
    __syncthreads();

    const int lane    = tid & 31;
    const int colBase = (tid >> 5) << 4;      // 8 waves -> 128 cols
    const int m       = lane & 15;
    const int kh      = (lane >> 4) << 1;     // 0 or 2
    const int col     = colBase + m;
    const float* brow = w + (size_t)col * IND;

    v8f c0 = {}, c1 = {};
    for (int ks = 0; ks < IND; ks += 4) {
        v2f a0, a1, b;
        b.x  = brow[ks + kh];
        b.y  = brow[ks + kh + 1];
        a0.x = aLDS[m * ASTRIDE + ks + kh];
        a0.y = aLDS[m * ASTRIDE + ks + kh + 1];
        a1.x = aLDS[(m + 16) * ASTRIDE + ks + kh];
        a1.y = aLDS[(m + 16) * ASTRIDE + ks + kh + 1];
        c0 = __builtin_amdgcn_wmma_f32_16x16x4_f32(false, a0, false, b,
                                                   (short)0, c0, false, false);
        c1 = __builtin_amdgcn_wmma_f32_16x16x4_f32(false, a1, false, b,
                                                   (short)0, c1, false, false);
    }
    const float bv = bias[col];
    const int rbase = (lane >> 4) << 3;       // 0 or 8
#pragma unroll
    for (int r = 0; r < 8; ++r) {
        int row0 = n0 + r + rbase;
        int row1 = row0 + 16;
        outf[(size_t)row0 * HID + col] = fmaxf(c0[r] + bv, 0.0f);
        outf[(size_t)row1 * HID + col] = fmaxf(c1[r] + bv, 0.0f);
    }
}

// ---------------------------------------------------------------------------
// One impeller layer, fully fused:
//   gather -> per-type weighted path sum -> per-type mean -> (32x256) fout in
//   LDS -> WMMA with fc_w^T (two A tiles per B fragment) -> relu -> residual.
// 1 block = 32 nodes, 256 threads. Gather: 8 threads/node x 16 H-cols each.
// ---------------------------------------------------------------------------
__global__ __launch_bounds__(256) void impeller_layer_kernel(
    const float* __restrict__ feats,      // (N,128) current (also feats_pre)
    const float* __restrict__ in_feats,   // (N,128)
    const int*   __restrict__ paths,      // (P,N,L)
    const int*   __restrict__ ptypes,     // (P)
    const float* __restrict__ pw,         // (2,8,128) this layer
    const float* __restrict__ fcw,        // (128,256) this layer
    float* __restrict__ feats_out)        // (N,128)
{
    __shared__ float wLDS[2 * PLEN * HID];        // 8 KB per-type weights
    __shared__ int   idxLDS[NPATH * TILE * PLEN]; // 8 KB path indices
    __shared__ int   ptLDS[NPATH];
    __shared__ float foutLDS[TILE * FSTRIDE];     // 33.3 KB fout tile

    const int tid = threadIdx.x;
    const int n0  = blockIdx.x * TILE;

    // warm L2 for the shared B matrix (128KB fc_w across 256 lanes)
    __builtin_prefetch(fcw + (size_t)tid * 128, 0, 0);

    for (int i = tid; i < 2 * PLEN * HID; i += 256) wLDS[i] = pw[i];
    if (tid < NPATH) ptLDS[tid] = ptypes[tid];
    for (int i = tid; i < NPATH * TILE * PLEN; i += 256) {
        int p = i >> 8, rem = i & 255;            // 256 = TILE*PLEN per path
        idxLDS[i] = paths[(size_t)p * NND * PLEN + (size_t)n0 * PLEN + rem];
    }
    __syncthreads();

    int cnt1 = 0;
#pragma unroll
    for (int p = 0; p < NPATH; ++p) cnt1 += ptLDS[p];
    const int cnt0 = NPATH - cnt1;
    const float inv1 = 1.0f / (float)(cnt1 > 0 ? cnt1 : 1);
    const float inv0 = 1.0f / (float)(cnt0 > 0 ? cnt0 : 1);

    const int node = tid >> 3;          // 0..31
    const int hb   = (tid & 7) << 4;    // 0..112 step 16

    float accA[16], acc1[16];
#pragma unroll
    for (int i = 0; i < 16; ++i) { accA[i] = 0.0f; acc1[i] = 0.0f; }

#pragma unroll
    for (int p = 0; p < NPATH; ++p) {
        const float tsel = (float)ptLDS[p];              // 0.0 or 1.0
        const float* wrow = &wLDS[ptLDS[p] * PLEN * HID];
#pragma unroll
        for (int l = 0; l < PLEN; ++l) {
            const int row = idxLDS[(p << 8) + (node << 3) + l];
            const float* fb = feats + (size_t)row * HID + hb;
            const float* wb = wrow + l * HID + hb;
#pragma unroll
            for (int q = 0; q < 4; ++q) {
                const float4 fv = *(const float4*)(fb + q * 4);
                const float4 wv = *(const float4*)(wb + q * 4);
                float v0 = fv.x * wv.x, v1 = fv.y * wv.y;
                float v2 = fv.z * wv.z, v3 = fv.w * wv.w;
                accA[q * 4 + 0] += v0; acc1[q * 4 + 0] = fmaf(tsel, v0, acc1[q * 4 + 0]);
                accA[q * 4 + 1] += v1; acc1[q * 4 + 1] = fmaf(tsel, v1, acc1[q * 4 + 1]);
                accA[q * 4 + 2] += v2; acc1[q * 4 + 2] = fmaf(tsel, v2, acc1[q * 4 + 2]);
                accA[q * 4 + 3] += v3; acc1[q * 4 + 3] = fmaf(tsel, v3, acc1[q * 4 + 3]);
            }
        }
    }
    // per-type means -> fout row (node), layout [type0 128 | type1 128]
#pragma unroll
    for (int i = 0; i < 16; ++i) {
        const float m1 = acc1[i] * inv1;
        const float m0 = (accA[i] - acc1[i]) * inv0;
        foutLDS[node * FSTRIDE + hb + i]       = m0;
        foutLDS[node * FSTRIDE + HID + hb + i] = m1;
    }
    __syncthreads();

    // GEMM: out(32x128) = fout(32x256) @ fcw^T; 8 waves, 2 tiles/wave sharing B
    const int lane    = tid & 31;
    const int colBase = (tid >> 5) << 4;
    const int m       = lane & 15;
    const int kh      = (lane >> 4) << 1;
    const int col     = colBase + m;
    const float* brow = fcw + (size_t)col * (2 * HID);

    v8f c0 = {}, c1 = {};
    for (int ks = 0; ks < 2 * HID; ks += 4) {
        v2f a0, a1, b;
        b.x  = brow[ks + kh];
        b.y  = brow[ks + kh + 1];
        a0.x = foutLDS[m * FSTRIDE + ks + kh];
        a0.y = foutLDS[m * FSTRIDE + ks + kh + 1];
        a1.x = foutLDS[(m + 16) * FSTRIDE + ks + kh];
        a1.y = foutLDS[(m + 16) * FSTRIDE + ks + kh + 1];
        c0 = __builtin_amdgcn_wmma_f32_16x16x4_f32(false, a0, false, b,
                                                   (short)0, c0, false, false);
        c1 = __builtin_amdgcn_wmma_f32_16x16x4_f32(false, a1, false, b,
                                                   (short)0, c1, false, false);
    }

    const int rbase = (lane >> 4) << 3;
#pragma unroll
    for (int r = 0; r < 8; ++r) {
        const int row0  = n0 + r + rbase;
        const int row1  = row0 + 16;
        const size_t i0 = (size_t)row0 * HID + col;
        const size_t i1 = (size_t)row1 * HID + col;
        const float v0  = fmaxf(c0[r], 0.0f);
        const float v1  = fmaxf(c1[r], 0.0f);
        feats_out[i0] = 0.8f * v0 + 0.1f * feats[i0] + 0.1f * in_feats[i0];
        feats_out[i1] = 0.8f * v1 + 0.1f * feats[i1] + 0.1f * in_feats[i1];
    }
}

// ---------------------------------------------------------------------------
// fc_out: out = relu(feats @ Wout^T + b)   feats:(N,128)  Wout:(64,128)
// 1 block = 32 rows x 64 cols, 4 waves, 2 tiles/wave sharing B.
// ---------------------------------------------------------------------------
__global__ __launch_bounds__(128) void fcout_kernel(
    const float* __restrict__ f, const float* __restrict__ w,
    const float* __restrict__ bias, float* __restrict__ out)
{
    __shared__ float aLDS[TILE * BSTRIDE];
    const int tid = threadIdx.x;
    const int n0  = blockIdx.x * TILE;

    for (int i = tid * 4; i < TILE * HID; i += 128 * 4) {
        int r = i >> 7, ccol = i & 127;
        *(float4*)&aLDS[r * BSTRIDE + ccol] =
            *(const float4*)&f[(size_t)(n0 + r) * HID + ccol];
    }
    __syncthreads();

    const int lane    = tid & 31;
    const int colBase = (tid >> 5) << 4;   // 4 waves -> 64 cols
    const int m       = lane & 15;
    const int kh      = (lane >> 4) << 1;
    const int col     = colBase + m;
    const float* brow = w + (size_t)col * HID;

    v8f c0 = {}, c1 = {};
    for (int ks = 0; ks < HID; ks += 4) {
        v2f a0, a1, b;
        b.x  = brow[ks + kh];
        b.y  = brow[ks + kh + 1];
        a0.x = aLDS[m * BSTRIDE + ks + kh];
        a0.y = aLDS[m * BSTRIDE + ks + kh + 1];
        a1.x = aLDS[(m + 16) * BSTRIDE + ks + kh];
        a1.y = aLDS[(m + 16) * BSTRIDE + ks + kh + 1];
        c0 = __builtin_amdgcn_wmma_f32_16x16x4_f32(false, a0, false, b,
                                                   (short)0, c0, false, false);
        c1 = __builtin_amdgcn_wmma_f32_16x16x4_f32(false, a1, false, b,
                                                   (short)0, c1, false, false);
    }
    const float bv = bias[col];
    const int rbase = (lane >> 4) << 3;
#pragma unroll
    for (int r = 0; r < 8; ++r) {
        int row0 = n0 + r + rbase;
        int row1 = row0 + 16;
        out[(size_t)row0 * OUTD + col] = fmaxf(c0[r] + bv, 0.0f);
        out[(size_t)row1 * OUTD + col] = fmaxf(c1[r] + bv, 0.0f);
    }
}

// ---------------------------------------------------------------------------
extern "C" void kernel_launch(void* const* d_in, const int* in_sizes, int n_in,
                              void* d_out, int out_size, void* d_ws, size_t ws_size,
                              hipStream_t stream)
{
    (void)in_sizes; (void)n_in; (void)out_size; (void)ws_size;
    const float* input_x    = (const float*)d_in[0];
    const int*   paths      = (const int*)  d_in[1];
    const int*   path_types = (const int*)  d_in[2];
    const float* fc_in_w    = (const float*)d_in[3];
    const float* fc_in_b    = (const float*)d_in[4];
    const float* fc_out_w   = (const float*)d_in[5];
    const float* fc_out_b   = (const float*)d_in[6];
    const float* layer_fc_w = (const float*)d_in[7];
    const float* path_w     = (const float*)d_in[8];
    float* out = (float*)d_out;

    const size_t featElems = (size_t)NND * HID;      // 2.56M floats = 10.24 MB
    float* in_feats = (float*)d_ws;                   // [0, 10.24MB)
    float* fA = in_feats + featElems;                 // ping
    float* fB = fA + featElems;                       // pong   (total ~30.7MB)

    dim3 grid(NND / TILE);                            // 625 blocks (exact)

    fcin_kernel<<<grid, 256, 0, stream>>>(input_x, fc_in_w, fc_in_b, in_feats);

    const float* src = in_feats;
    float* bufs[2] = { fA, fB };
    for (int i = 0; i < NLAY; ++i) {
        float* dst = bufs[i & 1];
        impeller_layer_kernel<<<grid, 256, 0, stream>>>(
            src, in_feats, paths, path_types,
            path_w     + (size_t)i * 2 * PLEN * HID,
            layer_fc_w + (size_t)i * HID * 2 * HID,
            dst);
        src = dst;
    }
    fcout_kernel<<<grid, 128, 0, stream>>>(src, fc_out_w, fc_out_b, out);
}